// HCPAttentionModel_71880572666598
// MI455X (gfx1250) — compile-verified
//
#include <hip/hip_runtime.h>
#include <hip/hip_bf16.h>

#define R_  100
#define B_  8
#define IW_ 64
#define OW_ 16
#define D_  128
#define BD_ 32

typedef __attribute__((ext_vector_type(2))) float v2f;
typedef __attribute__((ext_vector_type(8))) float v8f;

// ---------------------------------------------------------------------------
// Kernel 1: prologue. One block per region r (256 threads).
//   xn = (x - mean[r]) / (stddev[r]+1e-8)
//   q  = relu(xn @ q_in_W[r] + b) @ q_blk_W[r] + b          -> qws [R][B][D]
//   kv = relu(xn @ kv_in_W + b) @ kv_blk_W + b
//   K  = kv @ key_W + key_b                                  -> Kt [R][D][B] (transposed!)
//   V  = kv @ value_W + value_b ; Vn = V/max(||V||,1e-12)    -> Vn [B][R][BD]
// Threads 0..127 run the q path (column d), 128..255 run the kv path.
// ---------------------------------------------------------------------------
__global__ void __launch_bounds__(256) prologue_kernel(
    const float* __restrict__ x, const float* __restrict__ mean, const float* __restrict__ stddev,
    const float* __restrict__ kv_in_W, const float* __restrict__ kv_in_b,
    const float* __restrict__ kv_blk_W, const float* __restrict__ kv_blk_b,
    const float* __restrict__ q_in_W, const float* __restrict__ q_in_b,
    const float* __restrict__ q_blk_W, const float* __restrict__ q_blk_b,
    const float* __restrict__ key_W, const float* __restrict__ key_b,
    const float* __restrict__ value_W, const float* __restrict__ value_b,
    float* __restrict__ qws, float* __restrict__ Kt, float* __restrict__ Vn)
{
    const int r   = blockIdx.x;
    const int tid = threadIdx.x;

    __shared__ float sxn[B_][IW_];
    __shared__ float sq1[B_][D_];
    __shared__ float sk1[B_][D_];
    __shared__ float sk2[B_][D_];
    __shared__ float sV [B_][BD_];
    __shared__ float sInv[B_];

    const float mu  = mean[r];
    const float inv = 1.f / (stddev[r] + 1e-8f);
    for (int idx = tid; idx < B_ * IW_; idx += 256) {
        const int b = idx >> 6, i = idx & 63;
        sxn[b][i] = (x[((size_t)b * R_ + r) * IW_ + i] - mu) * inv;
    }
    __syncthreads();

    // ---- layer 1 ----
    if (tid < D_) {
        const int d = tid;
        float acc[B_];
        #pragma unroll
        for (int b = 0; b < B_; ++b) acc[b] = q_in_b[r * D_ + d];
        for (int i = 0; i < IW_; ++i) {
            const float w = q_in_W[((size_t)r * IW_ + i) * D_ + d];
            #pragma unroll
            for (int b = 0; b < B_; ++b) acc[b] += sxn[b][i] * w;
        }
        #pragma unroll
        for (int b = 0; b < B_; ++b) sq1[b][d] = fmaxf(acc[b], 0.f);
    } else {
        const int d = tid - D_;
        float acc[B_];
        #pragma unroll
        for (int b = 0; b < B_; ++b) acc[b] = kv_in_b[d];
        for (int i = 0; i < IW_; ++i) {
            const float w = kv_in_W[i * D_ + d];
            #pragma unroll
            for (int b = 0; b < B_; ++b) acc[b] += sxn[b][i] * w;
        }
        #pragma unroll
        for (int b = 0; b < B_; ++b) sk1[b][d] = fmaxf(acc[b], 0.f);
    }
    __syncthreads();

    // ---- layer 2 ----
    if (tid < D_) {
        const int d = tid;
        float acc[B_];
        #pragma unroll
        for (int b = 0; b < B_; ++b) acc[b] = q_blk_b[r * D_ + d];
        for (int e = 0; e < D_; ++e) {
            const float w = q_blk_W[((size_t)r * D_ + e) * D_ + d];
            #pragma unroll
            for (int b = 0; b < B_; ++b) acc[b] += sq1[b][e] * w;
        }
        #pragma unroll
        for (int b = 0; b < B_; ++b) qws[((size_t)r * B_ + b) * D_ + d] = acc[b];
    } else {
        const int d = tid - D_;
        float acc[B_];
        #pragma unroll
        for (int b = 0; b < B_; ++b) acc[b] = kv_blk_b[d];
        for (int e = 0; e < D_; ++e) {
            const float w = kv_blk_W[e * D_ + d];
            #pragma unroll
            for (int b = 0; b < B_; ++b) acc[b] += sk1[b][e] * w;
        }
        #pragma unroll
        for (int b = 0; b < B_; ++b) sk2[b][d] = acc[b];
    }
    __syncthreads();

    // ---- K (transposed store) and V ----
    if (tid < D_) {
        const int d = tid;
        float acc[B_];
        #pragma unroll
        for (int b = 0; b < B_; ++b) acc[b] = key_b[d];
        for (int e = 0; e < D_; ++e) {
            const float w = key_W[e * D_ + d];
            #pragma unroll
            for (int b = 0; b < B_; ++b) acc[b] += sk2[b][e] * w;
        }
        #pragma unroll
        for (int b = 0; b < B_; ++b) Kt[((size_t)r * D_ + d) * B_ + b] = acc[b];
    } else if (tid < D_ + BD_) {
        const int j = tid - D_;
        float acc[B_];
        #pragma unroll
        for (int b = 0; b < B_; ++b) acc[b] = value_b[j];
        for (int e = 0; e < D_; ++e) {
            const float w = value_W[e * BD_ + j];
            #pragma unroll
            for (int b = 0; b < B_; ++b) acc[b] += sk2[b][e] * w;
        }
        #pragma unroll
        for (int b = 0; b < B_; ++b) sV[b][j] = acc[b];
    }
    __syncthreads();
    if (tid < B_) {
        float s = 0.f;
        for (int j = 0; j < BD_; ++j) s += sV[tid][j] * sV[tid][j];
        sInv[tid] = 1.f / fmaxf(sqrtf(s), 1e-12f);
    }
    __syncthreads();
    {
        const int b = tid >> 5, j = tid & 31;
        Vn[((size_t)b * R_ + r) * BD_ + j] = sV[b][j] * sInv[b];
    }
}

// ---------------------------------------------------------------------------
// Kernel 2: fused bilinear attention logits (the HBM-streaming kernel).
//   attn_raw[b,s,t] = relu( sum_j (q_s W_st + bias_st)[b,j] * K[b,t,j] )
// Block = 8 waves, one wave per t; blockIdx.y = s, blockIdx.x = t-group.
// Per wave: y(8x128) = q_s(8x128) @ W_st(128x128) via V_WMMA_F32_16X16X4_F32,
// A from a zero-padded 16x132 LDS q-tile, B streamed from W with NT hints.
// A rolling global_prefetch (one instr = 32 lanes x 128B lines = 4KB, 8KB
// ahead of the consume point) supplies the in-flight bytes needed to saturate
// HBM without spending VGPRs or LOADcnt slots on deeper demand clauses.
// ---------------------------------------------------------------------------
__global__ void __launch_bounds__(256) attn_logits_kernel(
    const float* __restrict__ qws,   // [R][B][D]
    const float* __restrict__ Kt,    // [R][D][B]
    const float* __restrict__ qw,    // query_weight [R][R][D][D]
    const float* __restrict__ qbias, // query_bias   [R][R][D]
    float* __restrict__ attn_raw)    // [B][R][R]
{
    const int s    = blockIdx.y;
    const int wave = threadIdx.x >> 5;
    const int lane = threadIdx.x & 31;

    __shared__ float qt[16 * 132];   // rows = M (b, rows 8..15 zero), pitch 132

    for (int idx = threadIdx.x; idx < 16 * 132; idx += 256) {
        const int m = idx / 132, i = idx - m * 132;
        float v = 0.f;
        if (m < B_ && i < D_) v = qws[((size_t)s * B_ + m) * D_ + i];
        qt[idx] = v;
    }
    __syncthreads();

    const int t = blockIdx.x * 8 + wave;
    if (t >= R_) return;

    const int half = lane >> 4;      // lane/16
    const int l16  = lane & 15;      // lane%16

    const float* __restrict__ W = qw + ((size_t)s * R_ + t) * (D_ * D_);
    const float* arow = &qt[l16 * 132 + 2 * half];
    // each lane owns one 128B line per prefetch instruction: 32 lanes = 8 rows
    const float* pf = W + lane * 32;

    v8f c[8] = {};
    for (int k0 = 0; k0 < D_; k0 += 4) {
        // Rolling prefetch: rows [k0+16, k0+24), i.e. 8KB ahead of the demand
        // loads below. Rows 0..15 are covered by the initial demand clauses;
        // guard keeps every prefetch inside this wave's own 64KB tile.
        if (((k0 & 4) == 0) && (k0 <= 104)) {
            __builtin_prefetch(pf + (size_t)(k0 + 16) * D_, 0, 3);
        }
        // A fragment: A[m=l16, k=2*half+v] = q[b=l16, i=k0+2*half+v]
        v2f a = *(const v2f*)&arow[k0];
        const float* wk = W + (size_t)(k0 + 2 * half) * D_ + l16;
        #pragma unroll
        for (int nt = 0; nt < 8; ++nt) {
            // B fragment: B[k=2*half+v, n=l16] = W[k0+2*half+v, 16*nt+l16]
            v2f bf;
            bf.x = __builtin_nontemporal_load(wk + nt * 16);
            bf.y = __builtin_nontemporal_load(wk + nt * 16 + D_);
            c[nt] = __builtin_amdgcn_wmma_f32_16x16x4_f32(
                false, a, false, bf, (short)0, c[nt], false, false);
        }
    }

    // Epilogue: pb[b] = sum_j (y[b,j] + bias_j) * K[b,t,j]
    float pb[B_];
    #pragma unroll
    for (int b = 0; b < B_; ++b) pb[b] = 0.f;
    const float* __restrict__ bias = qbias + ((size_t)s * R_ + t) * D_;
    const float* __restrict__ ktp  = Kt + (size_t)t * D_ * B_;
    #pragma unroll
    for (int nt = 0; nt < 8; ++nt) {
        const int j = nt * 16 + l16;
        const float bj = bias[j];
        const float4 k0 = *(const float4*)(ktp + (size_t)j * B_);
        const float4 k1 = *(const float4*)(ktp + (size_t)j * B_ + 4);
        pb[0] += (c[nt][0] + bj) * k0.x;
        pb[1] += (c[nt][1] + bj) * k0.y;
        pb[2] += (c[nt][2] + bj) * k0.z;
        pb[3] += (c[nt][3] + bj) * k0.w;
        pb[4] += (c[nt][4] + bj) * k1.x;
        pb[5] += (c[nt][5] + bj) * k1.y;
        pb[6] += (c[nt][6] + bj) * k1.z;
        pb[7] += (c[nt][7] + bj) * k1.w;
    }
    // Reduce across the 16 N-lanes (masks <16 never cross the half boundary,
    // so the garbage rows M=8..15 in lanes 16..31 never contaminate lanes 0..15).
    #pragma unroll
    for (int m = 1; m < 16; m <<= 1) {
        #pragma unroll
        for (int b = 0; b < B_; ++b) pb[b] += __shfl_xor(pb[b], m, 32);
    }
    if (lane == 0) {
        #pragma unroll
        for (int b = 0; b < B_; ++b)
            attn_raw[((size_t)b * R_ + s) * R_ + t] = fmaxf(pb[b], 0.f);
    }
}

// ---------------------------------------------------------------------------
// Kernel 3: epilogue. One block per (s, b), 128 threads.
//   normalize attn over t, att = attn @ Vn, reproj, relu->post block, out proj,
//   rescale by stddev/mean.
// ---------------------------------------------------------------------------
__global__ void __launch_bounds__(128) epilogue_kernel(
    const float* __restrict__ attn_raw, const float* __restrict__ Vn,
    const float* __restrict__ reproj_W, const float* __restrict__ reproj_b,
    const float* __restrict__ post_W,   const float* __restrict__ post_b,
    const float* __restrict__ out_W,    const float* __restrict__ out_b,
    const float* __restrict__ mean,     const float* __restrict__ stddev,
    float* __restrict__ out)
{
    const int s = blockIdx.x, b = blockIdx.y;
    const int tid = threadIdx.x;

    __shared__ float red[128];
    __shared__ float an[R_];
    __shared__ float att[BD_];
    __shared__ float rr[D_];
    __shared__ float hh[D_];

    const float v = (tid < R_) ? attn_raw[((size_t)b * R_ + s) * R_ + tid] : 0.f;
    red[tid] = v;
    __syncthreads();
    for (int st = 64; st > 0; st >>= 1) {
        if (tid < st) red[tid] += red[tid + st];
        __syncthreads();
    }
    const float denom = 1e-8f + red[0];
    if (tid < R_) an[tid] = v / denom;
    __syncthreads();

    if (tid < BD_) {
        float acc = 0.f;
        for (int t = 0; t < R_; ++t)
            acc += an[t] * Vn[((size_t)b * R_ + t) * BD_ + tid];
        att[tid] = acc;
    }
    __syncthreads();

    {
        float acc = reproj_b[tid];
        for (int j = 0; j < BD_; ++j) acc += att[j] * reproj_W[j * D_ + tid];
        rr[tid] = fmaxf(acc, 0.f);   // relu(att_out) feeding the post block
    }
    __syncthreads();

    {
        float acc = post_b[s * D_ + tid];
        for (int d = 0; d < D_; ++d)
            acc += rr[d] * post_W[((size_t)s * D_ + d) * D_ + tid];
        hh[tid] = acc;
    }
    __syncthreads();

    if (tid < OW_) {
        float acc = out_b[s * OW_ + tid];
        for (int d = 0; d < D_; ++d)
            acc += hh[d] * out_W[((size_t)s * D_ + d) * OW_ + tid];
        out[((size_t)b * R_ + s) * OW_ + tid] = acc * stddev[s] + mean[s];
    }
}

// ---------------------------------------------------------------------------
extern "C" void kernel_launch(void* const* d_in, const int* in_sizes, int n_in,
                              void* d_out, int out_size, void* d_ws, size_t ws_size,
                              hipStream_t stream) {
    const float* x        = (const float*)d_in[0];
    const float* mean     = (const float*)d_in[1];
    const float* stddev   = (const float*)d_in[2];
    const float* kv_in_W  = (const float*)d_in[3];
    const float* kv_in_b  = (const float*)d_in[4];
    const float* kv_blk_W = (const float*)d_in[5];
    const float* kv_blk_b = (const float*)d_in[6];
    const float* q_in_W   = (const float*)d_in[7];
    const float* q_in_b   = (const float*)d_in[8];
    const float* q_blk_W  = (const float*)d_in[9];
    const float* q_blk_b  = (const float*)d_in[10];
    const float* key_W    = (const float*)d_in[11];
    const float* key_b    = (const float*)d_in[12];
    const float* value_W  = (const float*)d_in[13];
    const float* value_b  = (const float*)d_in[14];
    const float* query_W  = (const float*)d_in[15];
    const float* query_b  = (const float*)d_in[16];
    const float* reproj_W = (const float*)d_in[17];
    const float* reproj_b = (const float*)d_in[18];
    const float* post_W   = (const float*)d_in[19];
    const float* post_b   = (const float*)d_in[20];
    const float* out_W    = (const float*)d_in[21];
    const float* out_b    = (const float*)d_in[22];

    float* ws   = (float*)d_ws;
    float* qws  = ws;                 // R*B*D   = 102400
    float* Kt   = ws + 102400;        // R*D*B   = 102400
    float* Vn   = ws + 204800;        // B*R*BD  =  25600
    float* attn = ws + 230400;        // B*R*R   =  80000

    prologue_kernel<<<dim3(R_), 256, 0, stream>>>(
        x, mean, stddev, kv_in_W, kv_in_b, kv_blk_W, kv_blk_b,
        q_in_W, q_in_b, q_blk_W, q_blk_b, key_W, key_b, value_W, value_b,
        qws, Kt, Vn);

    attn_logits_kernel<<<dim3((R_ + 7) / 8, R_), 256, 0, stream>>>(
        qws, Kt, query_W, query_b, attn);

    epilogue_kernel<<<dim3(R_, B_), 128, 0, stream>>>(
        attn, Vn, reproj_W, reproj_b, post_W, post_b, out_W, out_b,
        mean, stddev, (float*)d_out);
}